// EGAT_79843442032709
// MI455X (gfx1250) — compile-verified
//
#include <hip/hip_runtime.h>
#include <hip/hip_bf16.h>
#include <math.h>

typedef float v2f __attribute__((ext_vector_type(2)));
typedef float v4f __attribute__((ext_vector_type(4)));
typedef float v8f __attribute__((ext_vector_type(8)));

#define AUGN 6
#define INF_ 128
#define ROWF 134   // IN + AUG
#define HEADS 8
#define HID 16
#define OUTF 40

// order-preserving float <-> uint for atomicMax-based segment max
__device__ __forceinline__ unsigned encf(float f) {
    unsigned u = __float_as_uint(f);
    return (u & 0x80000000u) ? ~u : (u | 0x80000000u);
}
__device__ __forceinline__ float decf(unsigned u) {
    return __uint_as_float((u & 0x80000000u) ? (u ^ 0x80000000u) : ~u);
}
#define ENC_NEG_INF 0x007FFFFFu   // encf(-inf)

__device__ __forceinline__ float lrelu02(float e) { return e > 0.f ? e : 0.2f * e; }

// ---------------- fill ----------------
__global__ void k_fill_u32(unsigned* p, unsigned v, long n) {
    long i = (long)blockIdx.x * blockDim.x + threadIdx.x;
    long st = (long)gridDim.x * blockDim.x;
    for (; i < n; i += st) p[i] = v;
}

// ---------------- aug column stats ----------------
__global__ void k_aug_stats(const float* __restrict__ x, float* __restrict__ stats, int N) {
    float s[AUGN] = {}, q[AUGN] = {};
    int st = gridDim.x * blockDim.x;
    for (int i = blockIdx.x * blockDim.x + threadIdx.x; i < N; i += st) {
        const float* r = x + (long)i * ROWF + INF_;
        #pragma unroll
        for (int c = 0; c < AUGN; c++) { float v = r[c]; s[c] += v; q[c] += v * v; }
    }
    #pragma unroll
    for (int c = 0; c < AUGN; c++) {
        float v = s[c], w = q[c];
        for (int off = 16; off; off >>= 1) {
            v += __shfl_xor(v, off, 32);
            w += __shfl_xor(w, off, 32);
        }
        if ((threadIdx.x & 31) == 0) {
            unsafeAtomicAdd(stats + c, v);
            unsafeAtomicAdd(stats + AUGN + c, w);
        }
    }
}

__global__ void k_aug_finalize(float* stats, int N) {
    if (blockIdx.x == 0 && threadIdx.x == 0) {
        #pragma unroll
        for (int c = 0; c < AUGN; c++) {
            float s = stats[c], q = stats[AUGN + c];
            float mean = s / (float)N;
            float var = (q - s * s / (float)N) / (float)(N - 1);
            stats[16 + c] = mean;
            stats[24 + c] = rsqrtf(var);
        }
    }
}

// ---------------- per-node gate: scale = 1 + sigmoid(MLP(aug_norm)) ----------------
__global__ void k_scale(const float* __restrict__ x, const float* __restrict__ stats,
                        const float* __restrict__ w1, const float* __restrict__ b1,
                        const float* __restrict__ w2, const float* __restrict__ b2,
                        float* __restrict__ scale, int N) {
    int n = blockIdx.x * blockDim.x + threadIdx.x;
    if (n >= N) return;
    const float* r = x + (long)n * ROWF + INF_;
    float a[AUGN];
    #pragma unroll
    for (int c = 0; c < AUGN; c++) a[c] = (r[c] - stats[16 + c]) * stats[24 + c];
    float acc = b2[0];
    #pragma unroll
    for (int j = 0; j < 16; j++) {
        float t = b1[j];
        #pragma unroll
        for (int c = 0; c < AUGN; c++) t += a[c] * w1[c * 16 + j];
        t = fmaxf(t, 0.f);
        acc += t * w2[j];
    }
    scale[n] = 1.f + 1.f / (1.f + expf(-acc));
}

// ---------------- GEMM1: h1 = (orig * scale) @ W1  (N x 128 @ 128 x 128), f32 WMMA ----------------
__global__ __launch_bounds__(256) void k_gemm1(const float* __restrict__ x,
                                               const float* __restrict__ scale,
                                               const float* __restrict__ W,
                                               float* __restrict__ h1, int N) {
    __shared__ float B[128 * 128];
    for (int i = threadIdx.x; i < 128 * 128 / 4; i += 256)
        ((v4f*)B)[i] = ((const v4f*)W)[i];
    __syncthreads();

    int wave = threadIdx.x >> 5, lane = threadIdx.x & 31;
    int tiles = N >> 4;                       // N multiple of 16 (50000)
    int tile = blockIdx.x * 8 + wave;
    if (tile >= tiles) return;
    int row0 = tile * 16;
    int half = lane >> 4, lr = lane & 15;
    int m = row0 + lr;
    float sc = scale[m];
    const float* arow = x + (long)m * ROWF;

    v8f acc[8] = {};
    for (int k0 = 0; k0 < 128; k0 += 4) {
        int kk = k0 + half * 2;
        v2f a;
        a.x = arow[kk] * sc;
        a.y = arow[kk + 1] * sc;
        #pragma unroll
        for (int t = 0; t < 8; t++) {
            v2f b;
            b.x = B[kk * 128 + t * 16 + lr];
            b.y = B[(kk + 1) * 128 + t * 16 + lr];
            acc[t] = __builtin_amdgcn_wmma_f32_16x16x4_f32(false, a, false, b,
                                                           (short)0, acc[t], false, false);
        }
    }
    #pragma unroll
    for (int t = 0; t < 8; t++)
        #pragma unroll
        for (int v = 0; v < 8; v++) {
            int mm = row0 + v + half * 8;
            h1[(long)mm * 128 + t * 16 + lr] = acc[t][v];
        }
}

// ---------------- attention coefficients layer 1 ----------------
__global__ void k_att1(const float* __restrict__ h1, const float* __restrict__ asw,
                       const float* __restrict__ adw, float* __restrict__ asrc,
                       float* __restrict__ adst, int N) {
    int idx = blockIdx.x * blockDim.x + threadIdx.x;
    if (idx >= N * HEADS) return;
    int n = idx >> 3, h = idx & 7;
    const float* r = h1 + (long)n * 128 + h * HID;
    float s = 0.f, d = 0.f;
    #pragma unroll
    for (int c = 0; c < HID; c++) { s += r[c] * asw[h * HID + c]; d += r[c] * adw[h * HID + c]; }
    asrc[(long)n * HEADS + h] = s;
    adst[(long)n * HEADS + h] = d;
}

// ---------------- edge passes layer 1 ----------------
__global__ void k_edge_max1(const int* __restrict__ ei, const float* __restrict__ asrc,
                            const float* __restrict__ adst, unsigned* __restrict__ emax,
                            int E, int EP) {
    int i = blockIdx.x * blockDim.x + threadIdx.x;
    if (i >= EP) return;
    int s, d;
    if (i < E) { s = ei[i]; d = ei[E + i]; } else { s = d = i - E; }
    const float* as = asrc + (long)s * HEADS;
    const float* ad = adst + (long)d * HEADS;
    #pragma unroll
    for (int h = 0; h < HEADS; h++) {
        float e = lrelu02(as[h] + ad[h]);
        atomicMax(emax + (long)d * HEADS + h, encf(e));
    }
}

__global__ void k_edge_sum1(const int* __restrict__ ei, const float* __restrict__ asrc,
                            const float* __restrict__ adst, const unsigned* __restrict__ emax,
                            float* __restrict__ den, int E, int EP) {
    int i = blockIdx.x * blockDim.x + threadIdx.x;
    if (i >= EP) return;
    int s, d;
    if (i < E) { s = ei[i]; d = ei[E + i]; } else { s = d = i - E; }
    const float* as = asrc + (long)s * HEADS;
    const float* ad = adst + (long)d * HEADS;
    #pragma unroll
    for (int h = 0; h < HEADS; h++) {
        float e = lrelu02(as[h] + ad[h]);
        float p = expf(e - decf(emax[(long)d * HEADS + h]));
        unsafeAtomicAdd(den + (long)d * HEADS + h, p);
    }
}

// one wave (32 lanes) per edge; lane covers 4 of the 128 features
__global__ void k_edge_agg1(const int* __restrict__ ei, const float* __restrict__ asrc,
                            const float* __restrict__ adst, const unsigned* __restrict__ emax,
                            const float* __restrict__ den, const float* __restrict__ h1,
                            float* __restrict__ out1, int E, int EP) {
    int gt = blockIdx.x * blockDim.x + threadIdx.x;
    int wi = gt >> 5, lane = gt & 31;
    if (wi >= EP) return;
    int s, d;
    if (wi < E) { s = ei[wi]; d = ei[E + wi]; } else { s = d = wi - E; }
    int h = lane >> 2;
    float e = lrelu02(asrc[(long)s * HEADS + h] + adst[(long)d * HEADS + h]);
    float alpha = expf(e - decf(emax[(long)d * HEADS + h])) / den[(long)d * HEADS + h];
    v4f hv = *(const v4f*)(h1 + (long)s * 128 + lane * 4);
    float* o = out1 + (long)d * 128 + lane * 4;
    unsafeAtomicAdd(o + 0, hv[0] * alpha);
    unsafeAtomicAdd(o + 1, hv[1] * alpha);
    unsafeAtomicAdd(o + 2, hv[2] * alpha);
    unsafeAtomicAdd(o + 3, hv[3] * alpha);
}

// ---------------- relu(out1 + bias1), in place ----------------
__global__ void k_relu_bias1(float* __restrict__ out1, const float* __restrict__ bias, long n4) {
    long i = (long)blockIdx.x * blockDim.x + threadIdx.x;
    if (i >= n4) return;
    v4f v = ((v4f*)out1)[i];
    int c = (int)((i & 31) * 4);
    v4f b = *(const v4f*)(bias + c);
    v4f r;
    #pragma unroll
    for (int j = 0; j < 4; j++) { float t = v[j] + b[j]; r[j] = t > 0.f ? t : 0.f; }
    ((v4f*)out1)[i] = r;
}

// ---------------- GEMM2: h2 = hrelu @ W2  (N x 128 @ 128 x 40), f32 WMMA ----------------
__global__ __launch_bounds__(256) void k_gemm2(const float* __restrict__ hrelu,
                                               const float* __restrict__ W,
                                               float* __restrict__ h2, int N) {
    __shared__ float B[128 * 48];   // padded 40 -> 48
    for (int i = threadIdx.x; i < 128 * 48; i += 256) {
        int k = i / 48, n = i % 48;
        B[i] = (n < OUTF) ? W[k * OUTF + n] : 0.f;
    }
    __syncthreads();

    int wave = threadIdx.x >> 5, lane = threadIdx.x & 31;
    int tiles = N >> 4;
    int tile = blockIdx.x * 8 + wave;
    if (tile >= tiles) return;
    int row0 = tile * 16;
    int half = lane >> 4, lr = lane & 15;
    int m = row0 + lr;
    const float* arow = hrelu + (long)m * 128;

    v8f acc[3] = {};
    for (int k0 = 0; k0 < 128; k0 += 4) {
        int kk = k0 + half * 2;
        v2f a;
        a.x = arow[kk];
        a.y = arow[kk + 1];
        #pragma unroll
        for (int t = 0; t < 3; t++) {
            v2f b;
            b.x = B[kk * 48 + t * 16 + lr];
            b.y = B[(kk + 1) * 48 + t * 16 + lr];
            acc[t] = __builtin_amdgcn_wmma_f32_16x16x4_f32(false, a, false, b,
                                                           (short)0, acc[t], false, false);
        }
    }
    #pragma unroll
    for (int t = 0; t < 3; t++)
        #pragma unroll
        for (int v = 0; v < 8; v++) {
            int mm = row0 + v + half * 8;
            int n = t * 16 + lr;
            if (n < OUTF) h2[(long)mm * OUTF + n] = acc[t][v];
        }
}

// ---------------- attention coefficients layer 2 (1 head, 40 ch) ----------------
__global__ void k_att2(const float* __restrict__ h2, const float* __restrict__ asw,
                       const float* __restrict__ adw, float* __restrict__ asrc,
                       float* __restrict__ adst, int N) {
    int n = blockIdx.x * blockDim.x + threadIdx.x;
    if (n >= N) return;
    const float* r = h2 + (long)n * OUTF;
    float s = 0.f, d = 0.f;
    #pragma unroll
    for (int c = 0; c < OUTF; c++) { s += r[c] * asw[c]; d += r[c] * adw[c]; }
    asrc[n] = s;
    adst[n] = d;
}

__global__ void k_edge_max2(const int* __restrict__ ei, const float* __restrict__ asrc,
                            const float* __restrict__ adst, unsigned* __restrict__ emax,
                            int E, int EP) {
    int i = blockIdx.x * blockDim.x + threadIdx.x;
    if (i >= EP) return;
    int s, d;
    if (i < E) { s = ei[i]; d = ei[E + i]; } else { s = d = i - E; }
    float e = lrelu02(asrc[s] + adst[d]);
    atomicMax(emax + d, encf(e));
}

__global__ void k_edge_sum2(const int* __restrict__ ei, const float* __restrict__ asrc,
                            const float* __restrict__ adst, const unsigned* __restrict__ emax,
                            float* __restrict__ den, int E, int EP) {
    int i = blockIdx.x * blockDim.x + threadIdx.x;
    if (i >= EP) return;
    int s, d;
    if (i < E) { s = ei[i]; d = ei[E + i]; } else { s = d = i - E; }
    float e = lrelu02(asrc[s] + adst[d]);
    unsafeAtomicAdd(den + d, expf(e - decf(emax[d])));
}

// one wave per edge; lanes cover the 40 outputs
__global__ void k_edge_agg2(const int* __restrict__ ei, const float* __restrict__ asrc,
                            const float* __restrict__ adst, const unsigned* __restrict__ emax,
                            const float* __restrict__ den, const float* __restrict__ h2,
                            float* __restrict__ out, int E, int EP) {
    int gt = blockIdx.x * blockDim.x + threadIdx.x;
    int wi = gt >> 5, lane = gt & 31;
    if (wi >= EP) return;
    int s, d;
    if (wi < E) { s = ei[wi]; d = ei[E + wi]; } else { s = d = wi - E; }
    float e = lrelu02(asrc[s] + adst[d]);
    float alpha = expf(e - decf(emax[d])) / den[d];
    for (int c = lane; c < OUTF; c += 32)
        unsafeAtomicAdd(out + (long)d * OUTF + c, h2[(long)s * OUTF + c] * alpha);
}

// ---------------- final: out = log_softmax(out + bias2), one wave per node, in place ----------------
__global__ void k_logsoftmax(float* __restrict__ out, const float* __restrict__ bias, int N) {
    int gt = blockIdx.x * blockDim.x + threadIdx.x;
    int n = gt >> 5, lane = gt & 31;
    if (n >= N) return;
    float* r = out + (long)n * OUTF;
    float v0 = r[lane] + bias[lane];                 // lane in [0,32) < 40, always valid
    bool has1 = lane < (OUTF - 32);
    float v1 = has1 ? (r[lane + 32] + bias[lane + 32]) : -INFINITY;
    float m = fmaxf(v0, v1);
    for (int off = 16; off; off >>= 1) m = fmaxf(m, __shfl_xor(m, off, 32));
    float sum = expf(v0 - m) + (has1 ? expf(v1 - m) : 0.f);
    for (int off = 16; off; off >>= 1) sum += __shfl_xor(sum, off, 32);
    float lse = logf(sum);
    r[lane] = v0 - m - lse;
    if (has1) r[lane + 32] = v1 - m - lse;
}

// ====================================================================
extern "C" void kernel_launch(void* const* d_in, const int* in_sizes, int n_in,
                              void* d_out, int out_size, void* d_ws, size_t ws_size,
                              hipStream_t stream) {
    const float* x        = (const float*)d_in[0];
    const int*   ei       = (const int*)d_in[1];
    const float* mlp_w1   = (const float*)d_in[2];
    const float* mlp_b1   = (const float*)d_in[3];
    const float* mlp_w2   = (const float*)d_in[4];
    const float* mlp_b2   = (const float*)d_in[5];
    const float* w1       = (const float*)d_in[6];
    const float* att_src1 = (const float*)d_in[7];
    const float* att_dst1 = (const float*)d_in[8];
    const float* bias1    = (const float*)d_in[9];
    const float* w2       = (const float*)d_in[10];
    const float* att_src2 = (const float*)d_in[11];
    const float* att_dst2 = (const float*)d_in[12];
    const float* bias2    = (const float*)d_in[13];
    float* out = (float*)d_out;

    const int N  = in_sizes[0] / ROWF;
    const int E  = in_sizes[1] / 2;
    const int EP = E + N;

    // workspace layout (float elements, 64-elem aligned regions)
    float* ws = (float*)d_ws;
    size_t o = 0;
    auto alloc = [&](size_t nn) { size_t r = o; o += (nn + 63) & ~(size_t)63; return r; };
    size_t o_stats = alloc(64);
    size_t o_scale = alloc((size_t)N);
    size_t o_h1    = alloc((size_t)N * 128);
    size_t o_as1   = alloc((size_t)N * HEADS);
    size_t o_ad1   = alloc((size_t)N * HEADS);
    size_t o_em1   = alloc((size_t)N * HEADS);
    size_t o_dn1   = alloc((size_t)N * HEADS);
    size_t o_out1  = alloc((size_t)N * 128);
    size_t o_h2    = alloc((size_t)N * OUTF);
    size_t o_as2   = alloc((size_t)N);
    size_t o_ad2   = alloc((size_t)N);
    size_t o_em2   = alloc((size_t)N);
    size_t o_dn2   = alloc((size_t)N);

    float*    stats = ws + o_stats;
    float*    scale = ws + o_scale;
    float*    h1    = ws + o_h1;
    float*    as1   = ws + o_as1;
    float*    ad1   = ws + o_ad1;
    unsigned* em1   = (unsigned*)(ws + o_em1);
    float*    dn1   = ws + o_dn1;
    float*    out1  = ws + o_out1;
    float*    h2    = ws + o_h2;
    float*    as2   = ws + o_as2;
    float*    ad2   = ws + o_ad2;
    unsigned* em2   = (unsigned*)(ws + o_em2);
    float*    dn2   = ws + o_dn2;

    const int B = 256;

    // init accumulators (every call: graph-replay safe)
    k_fill_u32<<<256, B, 0, stream>>>((unsigned*)stats, 0u, 64);
    k_fill_u32<<<1024, B, 0, stream>>>(em1, ENC_NEG_INF, (long)N * HEADS);
    k_fill_u32<<<1024, B, 0, stream>>>((unsigned*)dn1, 0u, (long)N * HEADS);
    k_fill_u32<<<2048, B, 0, stream>>>((unsigned*)out1, 0u, (long)N * 128);
    k_fill_u32<<<512, B, 0, stream>>>(em2, ENC_NEG_INF, (long)N);
    k_fill_u32<<<512, B, 0, stream>>>((unsigned*)dn2, 0u, (long)N);
    k_fill_u32<<<2048, B, 0, stream>>>((unsigned*)out, 0u, (long)N * OUTF);

    // gate
    k_aug_stats<<<128, B, 0, stream>>>(x, stats, N);
    k_aug_finalize<<<1, 32, 0, stream>>>(stats, N);
    k_scale<<<(N + B - 1) / B, B, 0, stream>>>(x, stats, mlp_w1, mlp_b1, mlp_w2, mlp_b2, scale, N);

    // layer 1
    int tiles = (N + 15) / 16;
    int gblocks = (tiles + 7) / 8;
    k_gemm1<<<gblocks, B, 0, stream>>>(x, scale, w1, h1, N);
    k_att1<<<((long)N * HEADS + B - 1) / B, B, 0, stream>>>(h1, att_src1, att_dst1, as1, ad1, N);
    k_edge_max1<<<(EP + B - 1) / B, B, 0, stream>>>(ei, as1, ad1, em1, E, EP);
    k_edge_sum1<<<(EP + B - 1) / B, B, 0, stream>>>(ei, as1, ad1, em1, dn1, E, EP);
    k_edge_agg1<<<(EP + 7) / 8, B, 0, stream>>>(ei, as1, ad1, em1, dn1, h1, out1, E, EP);
    k_relu_bias1<<<((long)N * 32 + B - 1) / B, B, 0, stream>>>(out1, bias1, (long)N * 32);

    // layer 2
    k_gemm2<<<gblocks, B, 0, stream>>>(out1, w2, h2, N);
    k_att2<<<(N + B - 1) / B, B, 0, stream>>>(h2, att_src2, att_dst2, as2, ad2, N);
    k_edge_max2<<<(EP + B - 1) / B, B, 0, stream>>>(ei, as2, ad2, em2, E, EP);
    k_edge_sum2<<<(EP + B - 1) / B, B, 0, stream>>>(ei, as2, ad2, em2, dn2, E, EP);
    k_edge_agg2<<<(EP + 7) / 8, B, 0, stream>>>(ei, as2, ad2, em2, dn2, h2, out, E, EP);

    // log-softmax (in place on d_out)
    k_logsoftmax<<<(N + 7) / 8, B, 0, stream>>>(out, bias2, N);
}